// RelationNetwork_33818572488893
// MI455X (gfx1250) — compile-verified
//
#include <hip/hip_runtime.h>
#include <hip/hip_bf16.h>

typedef __attribute__((ext_vector_type(2))) float v2f;
typedef __attribute__((ext_vector_type(8))) float v8f;
typedef __attribute__((ext_vector_type(4))) unsigned int uint32x4;
typedef __attribute__((ext_vector_type(8))) int int32x8;
typedef __attribute__((ext_vector_type(4))) int int32x4;

#define S_ROWS 320      // N_WAY * N_SUPPORT
#define Q_ROWS 1024
#define M_ROWS 1344     // S_ROWS + Q_ROWS
#define IN_DIM 2048
#define FEAT   512
#define HID    64
#define NWAY   20
#define NSUP   16

// ---------------------------------------------------------------------------
// Kernel 1: E[M_ROWS x FEAT] = [support_x ; query_x] @ enc_W + enc_b
// fp32 WMMA 16x16x4. One wave -> 16(M) x 64(N) tile, 4 waves per block.
// ---------------------------------------------------------------------------
__global__ __launch_bounds__(128)
void enc_gemm_kernel(const float* __restrict__ sx, const float* __restrict__ qx,
                     const float* __restrict__ W,  const float* __restrict__ bias,
                     float* __restrict__ E) {
    const int wave = threadIdx.x >> 5;
    const int lane = threadIdx.x & 31;
    const int tile  = blockIdx.x * 4 + wave;     // 0 .. 84*8-1
    const int mTile = tile >> 3;                 // 0..83
    const int n0    = (tile & 7) * 64;           // 0..448
    const int rowBase = mTile * 16;

    // 320 is a multiple of 16 -> each tile lives entirely in one source.
    const float* src = (rowBase < S_ROWS) ? (sx + (size_t)rowBase * IN_DIM)
                                          : (qx + (size_t)(rowBase - S_ROWS) * IN_DIM);

    const int mr = lane & 15;        // A: row in tile   / B: column in 16-wide subtile
    const int kh = (lane >> 4) * 2;  // K sub-offset: lanes 0-15 -> K+0,1 ; lanes 16-31 -> K+2,3
    const float* aRow = src + (size_t)mr * IN_DIM + kh;

    v8f acc0 = {}, acc1 = {}, acc2 = {}, acc3 = {};

    for (int k = 0; k < IN_DIM; k += 4) {
        // A fragment (16x4 f32): VGPR0 = K=kh, VGPR1 = K=kh+1 (contiguous b64 load)
        v2f a = *(const v2f*)(aRow + k);

        // B fragments (4x16 f32): lane holds column n, VGPR0/1 = rows k+kh, k+kh+1
        const float* bp = W + (size_t)(k + kh) * FEAT + n0 + mr;
        v2f b0, b1v, b2v, b3v;
        b0.x  = bp[0];   b0.y  = bp[FEAT];
        b1v.x = bp[16];  b1v.y = bp[FEAT + 16];
        b2v.x = bp[32];  b2v.y = bp[FEAT + 32];
        b3v.x = bp[48];  b3v.y = bp[FEAT + 48];

        acc0 = __builtin_amdgcn_wmma_f32_16x16x4_f32(false, a, false, b0,  (short)0, acc0, false, false);
        acc1 = __builtin_amdgcn_wmma_f32_16x16x4_f32(false, a, false, b1v, (short)0, acc1, false, false);
        acc2 = __builtin_amdgcn_wmma_f32_16x16x4_f32(false, a, false, b2v, (short)0, acc2, false, false);
        acc3 = __builtin_amdgcn_wmma_f32_16x16x4_f32(false, a, false, b3v, (short)0, acc3, false, false);
    }

    // Epilogue: C/D layout: VGPR i, lanes 0-15 -> M=i ; lanes 16-31 -> M=8+i
    const int erow = rowBase + (lane >> 4) * 8;
    v8f accs[4] = {acc0, acc1, acc2, acc3};
#pragma unroll
    for (int j = 0; j < 4; ++j) {
        const int col = n0 + j * 16 + mr;
        const float bv = bias[col];
#pragma unroll
        for (int i = 0; i < 8; ++i)
            E[(size_t)(erow + i) * FEAT + col] = accs[j][i] + bv;
    }
}

// ---------------------------------------------------------------------------
// Kernel 2: Hm[M_ROWS x HID] = E @ W1half ( + b1 folded into query rows ).
// Support rows use W1[0:512], query rows use W1[512:1024].
// ---------------------------------------------------------------------------
__global__ __launch_bounds__(128)
void head_gemm_kernel(const float* __restrict__ E, const float* __restrict__ W1,
                      const float* __restrict__ b1, float* __restrict__ Hm) {
    const int wave = threadIdx.x >> 5;
    const int lane = threadIdx.x & 31;
    const int tile = blockIdx.x * 4 + wave;      // 0..83
    const int rowBase = tile * 16;
    const bool isQ = (rowBase >= S_ROWS);        // uniform per wave (320 % 16 == 0)
    const float* Wp = W1 + (isQ ? (size_t)FEAT * HID : 0);

    const int mr = lane & 15;
    const int kh = (lane >> 4) * 2;
    const float* aRow = E + (size_t)(rowBase + mr) * FEAT + kh;

    v8f acc0 = {}, acc1 = {}, acc2 = {}, acc3 = {};

    for (int k = 0; k < FEAT; k += 4) {
        v2f a = *(const v2f*)(aRow + k);
        const float* bp = Wp + (size_t)(k + kh) * HID + mr;
        v2f b0, b1v, b2v, b3v;
        b0.x  = bp[0];   b0.y  = bp[HID];
        b1v.x = bp[16];  b1v.y = bp[HID + 16];
        b2v.x = bp[32];  b2v.y = bp[HID + 32];
        b3v.x = bp[48];  b3v.y = bp[HID + 48];

        acc0 = __builtin_amdgcn_wmma_f32_16x16x4_f32(false, a, false, b0,  (short)0, acc0, false, false);
        acc1 = __builtin_amdgcn_wmma_f32_16x16x4_f32(false, a, false, b1v, (short)0, acc1, false, false);
        acc2 = __builtin_amdgcn_wmma_f32_16x16x4_f32(false, a, false, b2v, (short)0, acc2, false, false);
        acc3 = __builtin_amdgcn_wmma_f32_16x16x4_f32(false, a, false, b3v, (short)0, acc3, false, false);
    }

    const int erow = rowBase + (lane >> 4) * 8;
    v8f accs[4] = {acc0, acc1, acc2, acc3};
#pragma unroll
    for (int j = 0; j < 4; ++j) {
        const int col = j * 16 + mr;
        const float bv = isQ ? b1[col] : 0.0f;   // fold b1 into hq only
#pragma unroll
        for (int i = 0; i < 8; ++i)
            Hm[(size_t)(erow + i) * HID + col] = accs[j][i] + bv;
    }
}

// ---------------------------------------------------------------------------
// Kernel 3 (fusion -- avoids the 84MB h tensor): per query q,
//   out[q][c] = mean_{j<16} sigmoid( relu(hq[q] + hs[16c+j]) . W2 + b2 ) / T
// The 320x64 fp32 support tile (80KB) is DMAed into LDS by the Tensor Data
// Mover (TENSOR_LOAD_TO_LDS, TENSORcnt), overlapping with per-thread hq loads.
// ---------------------------------------------------------------------------
__global__ __launch_bounds__(256)
void relation_kernel(const float* __restrict__ Hm, const float* __restrict__ W2,
                     const float* __restrict__ b2, const float* __restrict__ temp,
                     float* __restrict__ out) {
    extern __shared__ float smem[];          // S_ROWS*HID + HID floats
    float* hsL = smem;
    float* w2L = smem + S_ROWS * HID;

    const int tid = threadIdx.x;

    // --- TDM: DMA Hm[0:320, 0:64] (support h-matrix, contiguous 2D tile) into LDS.
    // One wave issues the descriptor; its TENSORcnt tracks completion.
    if (tid < 32) {
        const unsigned lds_off = (unsigned)(uintptr_t)hsL;               // LDS byte offset (low 32 bits)
        const unsigned long long ga = (unsigned long long)(uintptr_t)Hm; // tile start = tensor start

        // D# group 0: count=1 (user), lds_addr, global_addr[56:0], type=2 ("image")
        uint32x4 g0;
        g0[0] = 1u;                                    // count=1, is_restore=0, gather off
        g0[1] = lds_off;                               // bits [63:32]  lds_addr
        g0[2] = (unsigned)(ga & 0xFFFFFFFFu);          // bits [95:64]  global_addr[31:0]
        g0[3] = (unsigned)((ga >> 32) & 0x01FFFFFFu)   // bits [120:96] global_addr[56:32]
              | (2u << 30);                            // bits [127:126] type=2

        // D# group 1: data_size=4B, 2D tensor 64x320, tile = whole tensor
        int32x8 g1;
        g1[0] = (int)(2u << 16);         // workgroup_mask=0 (not in cluster), data_size=2 (4B)
        g1[1] = (int)(64u << 16);        // atomic_barrier_addr=0, tensor_dim0[15:0]=64
        g1[2] = (int)(320u << 16);       // tensor_dim0[31:16]=0, tensor_dim1[15:0]=320
        g1[3] = (int)(64u << 16);        // tensor_dim1[31:16]=0, tile_dim0=64
        g1[4] = (int)320u;               // tile_dim1=320, tile_dim2=0 (2D)
        g1[5] = (int)64u;                // tensor_dim0_stride[31:0]=64 elements
        g1[6] = 0;                       // tensor_dim0_stride[47:32]=0, tensor_dim1_stride lo=0
        g1[7] = 0;                       // tensor_dim1_stride=0 (unused, 2D tile)

        int32x4 g2 = {0, 0, 0, 0};       // unused (2D)
        int32x4 g3 = {0, 0, 0, 0};       // unused (2D)
        int32x8 g4 = {0, 0, 0, 0, 0, 0, 0, 0};  // extra operand (6-arg toolchain form)

        __builtin_amdgcn_tensor_load_to_lds(g0, g1, g2, g3, g4, 0);
    }

    if (tid < HID) w2L[tid] = W2[tid];

    // hq registers can load while the TDM runs (independent of LDS contents)
    const int q = blockIdx.x * 256 + tid;
    const float* hq = Hm + (size_t)(S_ROWS + q) * HID;   // b1 already folded in
    float hqv[HID];
#pragma unroll
    for (int h = 0; h < HID; ++h) hqv[h] = hq[h];

    const float bb  = b2[0];
    const float inv = 1.0f / (16.0f * temp[0]);

    if (tid < 32) __builtin_amdgcn_s_wait_tensorcnt(0);  // issuing wave drains TENSORcnt
    __syncthreads();                                     // publish LDS tile to all waves

    for (int c = 0; c < NWAY; ++c) {
        float csum = 0.0f;
        for (int j = 0; j < NSUP; ++j) {
            const float* hs = hsL + (c * NSUP + j) * HID;
            float r = bb;
#pragma unroll
            for (int h = 0; h < HID; ++h) {
                float t = hqv[h] + hs[h];                // broadcast LDS read
                r = __builtin_fmaf(fmaxf(t, 0.0f), w2L[h], r);
            }
            csum += 1.0f / (1.0f + __expf(-r));          // sigmoid
        }
        out[(size_t)q * NWAY + c] = csum * inv;
    }
}

// ---------------------------------------------------------------------------
extern "C" void kernel_launch(void* const* d_in, const int* in_sizes, int n_in,
                              void* d_out, int out_size, void* d_ws, size_t ws_size,
                              hipStream_t stream) {
    (void)in_sizes; (void)n_in; (void)out_size; (void)ws_size;
    const float* sx   = (const float*)d_in[0];   // support_x [320,2048]
    //            d_in[1] support_y : unused by reference (classes are contiguous)
    const float* qx   = (const float*)d_in[2];   // query_x   [1024,2048]
    //            d_in[3], d_in[4]: n_way / n_support scalars (compile-time here)
    const float* encW = (const float*)d_in[5];   // [2048,512]
    const float* encB = (const float*)d_in[6];   // [512]
    const float* W1   = (const float*)d_in[7];   // [1024,64]
    const float* b1   = (const float*)d_in[8];   // [64]
    const float* W2   = (const float*)d_in[9];   // [64]
    const float* b2   = (const float*)d_in[10];  // [1]
    const float* temp = (const float*)d_in[11];  // [1]
    float* out = (float*)d_out;

    float* E  = (float*)d_ws;                    // [1344,512]  (2.75 MB)
    float* Hm = E + (size_t)M_ROWS * FEAT;       // [1344,64]   (0.34 MB)

    // 84 M-tiles x 8 N-groups = 672 wave-tiles / 4 waves per block
    enc_gemm_kernel<<<168, 128, 0, stream>>>(sx, qx, encW, encB, E);
    // 84 wave-tiles / 4 waves per block
    head_gemm_kernel<<<21, 128, 0, stream>>>(E, W1, b1, Hm);

    const size_t shmem = (size_t)(S_ROWS * HID + HID) * sizeof(float); // ~80.25 KB
    relation_kernel<<<Q_ROWS / 256, 256, shmem, stream>>>(Hm, W2, b2, temp, out);
}